// GAT_86088324481664
// MI455X (gfx1250) — compile-verified
//
#include <hip/hip_runtime.h>
#include <hip/hip_bf16.h>
#include <math.h>

typedef __attribute__((ext_vector_type(2))) float v2f;
typedef __attribute__((ext_vector_type(8))) float v8f;

#define NEG_SLOPE 0.2f

// ---------------------------------------------------------------------------
// C[M,N] = A[M,K] * B[K,N], all row-major fp32. Requires M%16==0, N%16==0,
// K%4==0.  Block = (N/16) waves; wave w computes the 16x16 tile at
// (blockIdx.x*16, w*16) with a chain of V_WMMA_F32_16X16X4_F32.
// ---------------------------------------------------------------------------
__global__ void gemm_wmma_f32(const float* __restrict__ A,
                              const float* __restrict__ B,
                              float* __restrict__ C, int K, int N) {
  const int lane = threadIdx.x & 31;
  const int wave = threadIdx.x >> 5;
  const int m0   = blockIdx.x * 16;
  const int n0   = wave * 16;
  const int l15  = lane & 15;
  const int hi   = lane >> 4;            // 0: K pair {0,1}, 1: K pair {2,3}

  const float* ap = A + (size_t)(m0 + l15) * K + hi * 2;   // A row, K offset
  const float* bp = B + (size_t)(hi * 2) * N + (n0 + l15); // B K row, column

  v8f acc = {};
  for (int k0 = 0; k0 < K; k0 += 4) {
    v2f a, b;
    a.x = ap[0];
    a.y = ap[1];
    b.x = bp[0];
    b.y = bp[N];
    ap += 4;
    bp += (size_t)4 * N;
    acc = __builtin_amdgcn_wmma_f32_16x16x4_f32(
        /*neg_a=*/false, a, /*neg_b=*/false, b,
        /*c_mod=*/(short)0, acc, /*reuse_a=*/false, /*reuse_b=*/false);
  }

  // D layout: lanes 0-15 rows m0+0..7, lanes 16-31 rows m0+8..15, col n0+l15
  float* cp = C + (size_t)(m0 + hi * 8) * N + (n0 + l15);
#pragma unroll
  for (int r = 0; r < 8; ++r) cp[(size_t)r * N] = acc[r];
}

// ---------------------------------------------------------------------------
// alpha_src[n,h] = sum_c h[n,h,c]*a_src[h,c]   (same for dst)
// ---------------------------------------------------------------------------
__global__ void alpha_kernel(const float* __restrict__ h,
                             const float* __restrict__ a_src,
                             const float* __restrict__ a_dst,
                             float* __restrict__ alpha_src,
                             float* __restrict__ alpha_dst,
                             int NH, int H, int C) {
  int i = blockIdx.x * blockDim.x + threadIdx.x;
  if (i >= NH) return;
  int hd = i % H;
  const float* hp = h + (size_t)i * C;          // i == n*H + hd, h row-major
  const float* as = a_src + hd * C;
  const float* ad = a_dst + hd * C;
  float s = 0.f, d = 0.f;
  for (int c = 0; c < C; ++c) {
    float v = hp[c];
    s += v * as[c];
    d += v * ad[c];
  }
  alpha_src[i] = s;
  alpha_dst[i] = d;
}

__global__ void fill_kernel(float* __restrict__ p, float v, int n) {
  int i = blockIdx.x * blockDim.x + threadIdx.x;
  if (i < n) p[i] = v;
}

// Monotone fp32 atomic-max via integer ordering tricks (handles negatives).
__device__ inline void atomicMaxF32(float* addr, float val) {
  if (val >= 0.f)
    atomicMax((int*)addr, __float_as_int(val));
  else
    atomicMin((unsigned int*)addr, __float_as_uint(val));
}

// Pass 1: e = leaky_relu(asrc[src]+adst[dst]); store raw e; segment max on dst
__global__ void edge_lrelu_max(const int* __restrict__ src,
                               const int* __restrict__ dst,
                               const float* __restrict__ asrc,
                               const float* __restrict__ adst,
                               float* __restrict__ emax,
                               float* __restrict__ ee, int E, int H) {
  int e = blockIdx.x * blockDim.x + threadIdx.x;
  if (e >= E) return;
  int s = src[e], d = dst[e];
  for (int h = 0; h < H; ++h) {
    float v = asrc[s * H + h] + adst[d * H + h];
    v = (v > 0.f) ? v : NEG_SLOPE * v;
    ee[(size_t)e * H + h] = v;
    atomicMaxF32(&emax[d * H + h], v);
  }
}

__global__ void fix_emax(float* __restrict__ emax, int n) {
  int i = blockIdx.x * blockDim.x + threadIdx.x;
  if (i >= n) return;
  float v = emax[i];
  emax[i] = isfinite(v) ? v : 0.f;
}

// Pass 2: ee = exp(e - emax[dst]); segment sum into denom
__global__ void edge_exp_sum(const int* __restrict__ dst,
                             const float* __restrict__ emax,
                             float* __restrict__ ee,
                             float* __restrict__ denom, int E, int H) {
  int e = blockIdx.x * blockDim.x + threadIdx.x;
  if (e >= E) return;
  int d = dst[e];
  for (int h = 0; h < H; ++h) {
    float x = expf(ee[(size_t)e * H + h] - emax[d * H + h]);
    ee[(size_t)e * H + h] = x;
    atomicAdd(&denom[d * H + h], x);
  }
}

// Pass 3: out[dst] += h[src] * (ee / (denom[dst]+1e-16)).  One wave per edge,
// lanes stride channels (HC = H<<cshift).
__global__ void edge_aggregate(const int* __restrict__ src,
                               const int* __restrict__ dst,
                               const float* __restrict__ h,
                               const float* __restrict__ ee,
                               const float* __restrict__ denom,
                               float* __restrict__ out, int E, int H,
                               int cshift) {
  int wid  = (blockIdx.x * blockDim.x + threadIdx.x) >> 5;
  int lane = threadIdx.x & 31;
  if (wid >= E) return;
  int s = src[wid], d = dst[wid];
  const int HC = H << cshift;
  const float* hrow = h + (size_t)s * HC;
  float* orow = out + (size_t)d * HC;
  __builtin_prefetch(hrow, 0, 0);
  for (int c = lane; c < HC; c += 32) {
    int hd = c >> cshift;
    float alpha =
        ee[(size_t)wid * H + hd] / (denom[d * H + hd] + 1e-16f);
    atomicAdd(&orow[c], hrow[c] * alpha);
  }
}

__global__ void bias_relu(float* __restrict__ o, const float* __restrict__ b,
                          int n, int F) {
  int i = blockIdx.x * blockDim.x + threadIdx.x;
  if (i >= n) return;
  float v = o[i] + b[i % F];
  o[i] = v > 0.f ? v : 0.f;
}

__global__ void bias_add(float* __restrict__ o, const float* __restrict__ b,
                         int n, int F) {
  int i = blockIdx.x * blockDim.x + threadIdx.x;
  if (i >= n) return;
  o[i] = o[i] + b[i % F];
}

// ---------------------------------------------------------------------------
extern "C" void kernel_launch(void* const* d_in, const int* in_sizes, int n_in,
                              void* d_out, int out_size, void* d_ws,
                              size_t ws_size, hipStream_t stream) {
  const float* x    = (const float*)d_in[0];
  const int*   ei   = (const int*)d_in[1];
  const float* W1   = (const float*)d_in[2];
  const float* as1w = (const float*)d_in[3];
  const float* ad1w = (const float*)d_in[4];
  const float* b1   = (const float*)d_in[5];
  const float* W2   = (const float*)d_in[6];
  const float* as2w = (const float*)d_in[7];
  const float* ad2w = (const float*)d_in[8];
  const float* b2   = (const float*)d_in[9];

  const int N = in_sizes[0] / 128;  // 50000 (divisible by 16)
  const int E = in_sizes[1] / 2;    // 640000
  const int* src = ei;
  const int* dst = ei + E;
  float* out = (float*)d_out;

  // workspace carve-up (fp32)
  float* p   = (float*)d_ws;
  float* h1  = p; p += (size_t)N * 128;
  float* o1  = p; p += (size_t)N * 128;
  float* as1 = p; p += (size_t)N * 2;
  float* ad1 = p; p += (size_t)N * 2;
  float* em1 = p; p += (size_t)N * 2;
  float* dn1 = p; p += (size_t)N * 2;
  float* ee1 = p; p += (size_t)E * 2;
  float* h2  = p; p += (size_t)N * 32;
  float* as2 = p; p += (size_t)N;
  float* ad2 = p; p += (size_t)N;
  float* em2 = p; p += (size_t)N;
  float* dn2 = p; p += (size_t)N;
  float* ee2 = p; p += (size_t)E;

  const int T = 256;
  auto cdiv = [](int a, int b) { return (a + b - 1) / b; };

  // ---------------- Layer 1 (H=2, C=64, concat) ----------------
  gemm_wmma_f32<<<N / 16, 256, 0, stream>>>(x, W1, h1, 128, 128);
  alpha_kernel<<<cdiv(N * 2, T), T, 0, stream>>>(h1, as1w, ad1w, as1, ad1,
                                                 N * 2, 2, 64);
  fill_kernel<<<cdiv(N * 2, T), T, 0, stream>>>(em1, -INFINITY, N * 2);
  fill_kernel<<<cdiv(N * 2, T), T, 0, stream>>>(dn1, 0.f, N * 2);
  fill_kernel<<<cdiv(N * 128, T), T, 0, stream>>>(o1, 0.f, N * 128);
  edge_lrelu_max<<<cdiv(E, T), T, 0, stream>>>(src, dst, as1, ad1, em1, ee1,
                                               E, 2);
  fix_emax<<<cdiv(N * 2, T), T, 0, stream>>>(em1, N * 2);
  edge_exp_sum<<<cdiv(E, T), T, 0, stream>>>(dst, em1, ee1, dn1, E, 2);
  edge_aggregate<<<cdiv(E, 8), T, 0, stream>>>(src, dst, h1, ee1, dn1, o1, E,
                                               2, /*cshift=*/6);
  bias_relu<<<cdiv(N * 128, T), T, 0, stream>>>(o1, b1, N * 128, 128);

  // ---------------- Layer 2 (H=1, C=32, mean==identity) ----------------
  gemm_wmma_f32<<<N / 16, 64, 0, stream>>>(o1, W2, h2, 128, 32);
  alpha_kernel<<<cdiv(N, T), T, 0, stream>>>(h2, as2w, ad2w, as2, ad2, N, 1,
                                             32);
  fill_kernel<<<cdiv(N, T), T, 0, stream>>>(em2, -INFINITY, N);
  fill_kernel<<<cdiv(N, T), T, 0, stream>>>(dn2, 0.f, N);
  fill_kernel<<<cdiv(N * 32, T), T, 0, stream>>>(out, 0.f, N * 32);
  edge_lrelu_max<<<cdiv(E, T), T, 0, stream>>>(src, dst, as2, ad2, em2, ee2,
                                               E, 1);
  fix_emax<<<cdiv(N, T), T, 0, stream>>>(em2, N);
  edge_exp_sum<<<cdiv(E, T), T, 0, stream>>>(dst, em2, ee2, dn2, E, 1);
  edge_aggregate<<<cdiv(E, 8), T, 0, stream>>>(src, dst, h2, ee2, dn2, out, E,
                                               1, /*cshift=*/5);
  bias_add<<<cdiv(N * 32, T), T, 0, stream>>>(out, b2, N * 32, 32);
}